// OctreeInterp_24146306138646
// MI455X (gfx1250) — compile-verified
//
#include <hip/hip_runtime.h>
#include <stdint.h>

// ---------------------------------------------------------------------------
// Octree trilinear interpolation for MI455X (gfx1250, wave32).
// Memory-bound gather kernel: transpose data to (H,C), then async-stage
// 8x256B corner rows per point into LDS (CDNA5 GLOBAL_LOAD_ASYNC_TO_LDS),
// reduce with per-lane FMAs, write coalesced (C,N) output via LDS re-tile.
// ---------------------------------------------------------------------------

#ifndef __has_builtin
#define __has_builtin(x) 0
#endif

#if __has_builtin(__builtin_amdgcn_global_load_async_to_lds_b64)
#define HAVE_ASYNC_LDS 1
#else
#define HAVE_ASYNC_LDS 0
#endif

// Builtin signature (probe-confirmed on this toolchain):
//   param0: int __vector_size__(8) addrspace(1)*  (global source)
//   param1: int __vector_size__(8) addrspace(3)*  (LDS destination)
//   param2/3: immediate int offset / cpol
typedef int v2i __attribute__((vector_size(8)));
typedef __attribute__((address_space(1))) v2i* gptr_v2i;
typedef __attribute__((address_space(3))) v2i* lptr_v2i;

__device__ __forceinline__ void compiler_fence() { asm volatile("" ::: "memory"); }

__device__ __forceinline__ void wait_async_le8() {
#if __has_builtin(__builtin_amdgcn_s_wait_asynccnt)
  compiler_fence();
  __builtin_amdgcn_s_wait_asynccnt(8);
  compiler_fence();
#else
  asm volatile("s_wait_asynccnt 0x8" ::: "memory");
#endif
}

__device__ __forceinline__ void wait_async_le0() {
#if __has_builtin(__builtin_amdgcn_s_wait_asynccnt)
  compiler_fence();
  __builtin_amdgcn_s_wait_asynccnt(0);
  compiler_fence();
#else
  asm volatile("s_wait_asynccnt 0x0" ::: "memory");
#endif
}

// ---------------------------------------------------------------------------
// Kernel 1: transpose data (C=64, H) channel-major -> feat (H, 64) row-major.
// Full 64x64 tiles take a uniform-branch fast path with b128 global loads and
// b128 global stores (768MB of the ~1.35GB total traffic lives here, so VMEM
// instruction efficiency matters). LDS patterns are bank-conflict-free.
// ---------------------------------------------------------------------------
__global__ __launch_bounds__(256) void oct_transpose_ch_hc(
    const float* __restrict__ data, float* __restrict__ feat, int H) {
  __shared__ float tile[64][65];  // +1 pad
  const int tid = threadIdx.x;
  const int h0 = blockIdx.x * 64;

  if (h0 + 64 <= H) {
    // ---- fast path: full tile, vectorized
    const int hq = (tid & 15) * 4;  // h sub-offset 0..60 step 4
    const int cq = tid >> 4;        // c base 0..15
#pragma unroll
    for (int r = 0; r < 4; ++r) {
      const int c = cq + 16 * r;
      const float4 v = *(const float4*)(data + (size_t)c * H + h0 + hq);
      tile[hq + 0][c] = v.x;
      tile[hq + 1][c] = v.y;
      tile[hq + 2][c] = v.z;
      tile[hq + 3][c] = v.w;
    }
    __syncthreads();
    const int cc = tid & 15;   // channel group (4 channels)
    const int hb = tid >> 4;   // h sub-index 0..15
#pragma unroll
    for (int r = 0; r < 4; ++r) {
      const int hl = hb + 16 * r;
      float4 v;
      v.x = tile[hl][4 * cc + 0];
      v.y = tile[hl][4 * cc + 1];
      v.z = tile[hl][4 * cc + 2];
      v.w = tile[hl][4 * cc + 3];
      *(float4*)(feat + (size_t)(h0 + hl) * 64 + 4 * cc) = v;
    }
  } else {
    // ---- tail path: element-guarded (only the last block)
    const int a = tid & 63;
    const int q = tid >> 6;
#pragma unroll
    for (int r = 0; r < 16; ++r) {
      const int c = r * 4 + q;
      const int h = h0 + a;
      tile[a][c] = (h < H) ? data[(size_t)c * H + h] : 0.0f;
    }
    __syncthreads();
#pragma unroll
    for (int r = 0; r < 16; ++r) {
      const int h = h0 + r * 4 + q;
      if (h < H) feat[(size_t)h * 64 + a] = tile[r * 4 + q][a];
    }
  }
}

// ---------------------------------------------------------------------------
// Kernel 2: interpolation. MODE 0: feat is (H,64) transposed (fast path,
// async-LDS staged gathers). MODE 1: feat is original (64,H) strided.
// Block = 256 threads (8 wave32), 32 points per block.
// ---------------------------------------------------------------------------
template <int MODE>
__global__ __launch_bounds__(256) void oct_interp(
    const float* __restrict__ feat, const float* __restrict__ pts,
    const long long* __restrict__ keys, float* __restrict__ out, int H, int N) {
  __shared__ int pos_s[256];    // [32 pts][8 corners]
  __shared__ float w_s[256];
  __shared__ __align__(16)
      float stage[(MODE == 0 && HAVE_ASYNC_LDS) ? 8 * 2 * 8 * 64 : 1];
  __shared__ __align__(8) float outs[32 * 66];  // [pt][ch], pitch 66

  const int tid = threadIdx.x;
  const int n0 = blockIdx.x * 32;

  // ---- Phase A: one thread per (point, corner): key, weight, binary search
  {
    const int p = tid >> 3;
    const int k = tid & 7;
    const int n = n0 + p;
    const int nc = (n < N) ? n : (N - 1);
    const float px = pts[4 * nc + 0];
    const float py = pts[4 * nc + 1];
    const float pz = pts[4 * nc + 2];
    const long long id = (long long)pts[4 * nc + 3];
    const float xf = (px + 1.0f) * 64.0f - 0.5f;  // 2^(DEPTH-1)=64
    const float yf = (py + 1.0f) * 64.0f - 0.5f;
    const float zf = (pz + 1.0f) * 64.0f - 0.5f;
    const float xb = floorf(xf), yb = floorf(yf), zb = floorf(zf);
    const float fx = xf - xb, fy = yf - yb, fz = zf - zb;
    const long long M = 0xFFFFll;
    long long key = ((id & M) << 48) | (((long long)xb & M) << 32) |
                    (((long long)yb & M) << 16) | ((long long)zb & M);
    // MASKU pairing from the reference: bit k&1 -> +x field, (k>>1)&1 -> +y,
    // (k>>2)&1 -> +z field of the packed key.
    key += ((long long)(k & 1) << 32) | ((long long)((k >> 1) & 1) << 16) |
           (long long)((k >> 2) & 1);
    const float m0 = (float)((k >> 2) & 1);
    const float m1 = (float)((k >> 1) & 1);
    const float m2 = (float)(k & 1);
    float w = fabsf(((1.0f - m0) - fx) * ((1.0f - m1) - fy) * ((1.0f - m2) - fz));
    // lower_bound over sorted keys (12MB -> lives in the 192MB L2)
    int lo = 0, hi = H;
    while (lo < hi) {
      const int mid = (lo + hi) >> 1;
      if (keys[mid] < key) lo = mid + 1; else hi = mid;
    }
    const int posc = (lo < H - 1) ? lo : (H - 1);
    if (keys[posc] != key || n >= N) w = 0.0f;
    pos_s[tid] = posc;
    w_s[tid] = w;
  }
  __syncthreads();

  // ---- Phase B: wave-per-4-points; 2 channels per lane
  const int wave = tid >> 5;
  const int lane = tid & 31;
  const int pg0 = wave * 4;

#if HAVE_ASYNC_LDS
  auto issue = [&](int buf, int pg) {
    if (MODE == 0) {
#pragma unroll
      for (int k = 0; k < 8; ++k) {
        const float* g = feat + (size_t)pos_s[pg * 8 + k] * 64 + lane * 2;
        float* l = &stage[((wave * 2 + buf) * 8 + k) * 64 + lane * 2];
        __builtin_amdgcn_global_load_async_to_lds_b64((gptr_v2i)g, (lptr_v2i)l,
                                                      0, 0);
      }
    }
  };
  if (MODE == 0) issue(0, pg0);
#endif

  for (int p = 0; p < 4; ++p) {
    const int pg = pg0 + p;
#if HAVE_ASYNC_LDS
    if (MODE == 0) {
      if (p < 3) { issue((p + 1) & 1, pg + 1); wait_async_le8(); }
      else       { wait_async_le0(); }
    }
#endif
    float ax = 0.0f, ay = 0.0f, norm = 0.0f;
#pragma unroll
    for (int k = 0; k < 8; ++k) {
      const float wk = w_s[pg * 8 + k];
      norm += wk;
      float vx, vy;
      if (MODE == 0) {
#if HAVE_ASYNC_LDS
        const float2 v =
            *(const float2*)&stage[((wave * 2 + (p & 1)) * 8 + k) * 64 + lane * 2];
        vx = v.x; vy = v.y;
#else
        const float2 v =
            *(const float2*)(feat + (size_t)pos_s[pg * 8 + k] * 64 + lane * 2);
        vx = v.x; vy = v.y;
#endif
      } else {
        const size_t h = (size_t)pos_s[pg * 8 + k];
        vx = feat[(size_t)(2 * lane) * H + h];
        vy = feat[(size_t)(2 * lane + 1) * H + h];
      }
      ax = fmaf(wk, vx, ax);
      ay = fmaf(wk, vy, ay);
    }
    const float inv = 1.0f / (norm + 1e-10f);
    float2 res;
    res.x = ax * inv;
    res.y = ay * inv;
    *(float2*)&outs[pg * 66 + lane * 2] = res;  // pg*66+2*lane is even: 8B ok
  }
  __syncthreads();

  // ---- coalesced (C, N) store: lanes sweep 32 consecutive n per channel
#pragma unroll
  for (int i = 0; i < 8; ++i) {
    const int idx = i * 256 + tid;
    const int c = idx >> 5;
    const int p = idx & 31;
    const int n = n0 + p;
    if (n < N) out[(size_t)c * N + n] = outs[p * 66 + c];
  }
}

// ---------------------------------------------------------------------------
extern "C" void kernel_launch(void* const* d_in, const int* in_sizes, int n_in,
                              void* d_out, int out_size, void* d_ws,
                              size_t ws_size, hipStream_t stream) {
  const float* data = (const float*)d_in[0];          // (1, 64, H, 1)
  const float* pts = (const float*)d_in[1];           // (N, 4)
  const long long* keys = (const long long*)d_in[2];  // (H,)
  float* out = (float*)d_out;                         // (1, 64, N, 1)
  const int H = in_sizes[2];
  const int N = in_sizes[1] / 4;
  const size_t need = (size_t)H * 64 * sizeof(float);
  const int iblocks = (N + 31) / 32;
  if (ws_size >= need) {
    float* feat = (float*)d_ws;
    const int tblocks = (H + 63) / 64;
    oct_transpose_ch_hc<<<tblocks, 256, 0, stream>>>(data, feat, H);
    oct_interp<0><<<iblocks, 256, 0, stream>>>(feat, pts, keys, out, H, N);
  } else {
    oct_interp<1><<<iblocks, 256, 0, stream>>>(data, pts, keys, out, H, N);
  }
}